// SEQMultiHeadAttention_3341484556825
// MI455X (gfx1250) — compile-verified
//
#include <hip/hip_runtime.h>

typedef __attribute__((ext_vector_type(2))) float v2f;
typedef __attribute__((ext_vector_type(8))) float v8f;

#define NEG_INF_F (-1000000000.0f)

// ---------------------------------------------------------------------------
// C(32x512) = A(32x512) @ W(512x512) * scale, via V_WMMA_F32_16X16X4_F32.
// One wave32 computes one 16x16 tile of C, K accumulated in 128 steps of 4.
// Block = 256 threads (8 waves -> 8 N-tiles); grid = (512/128, 32/16).
// Fragment layouts per CDNA5 ISA 7.12.2:
//   A 16x4 : lane%16 = M, K = 2*(lane/16) + vgpr
//   B 4x16 : lane%16 = N, K = 2*(lane/16) + vgpr
//   C 16x16: lane%16 = N, M = 8*(lane/16) + vgpr
// ---------------------------------------------------------------------------
__global__ __launch_bounds__(256) void gemm32_wmma(const float* __restrict__ A,
                                                   const float* __restrict__ W,
                                                   float* __restrict__ C,
                                                   float scale)
{
    constexpr int N = 512, K = 512;
    const int wave  = threadIdx.x >> 5;
    const int lane  = threadIdx.x & 31;
    const int lg    = lane >> 4;     // lane group: 0 (lanes 0-15) or 1 (16-31)
    const int lm    = lane & 15;
    const int nBase = (blockIdx.x * 8 + wave) * 16;
    const int mBase = blockIdx.y * 16;

    v8f acc = {};
    const float* arow = A + (size_t)(mBase + lm) * K + 2 * lg;
    const float* bcol = W + (size_t)(2 * lg) * N + nBase + lm;

    for (int kb = 0; kb < K; kb += 4) {
        v2f a;
        a.x = arow[kb];
        a.y = arow[kb + 1];
        v2f bf;
        const float* bp = bcol + (size_t)kb * N;
        bf.x = bp[0];
        bf.y = bp[N];
        acc = __builtin_amdgcn_wmma_f32_16x16x4_f32(
            /*neg_a=*/false, a, /*neg_b=*/false, bf,
            /*c_mod=*/(short)0, acc, /*reuse_a=*/false, /*reuse_b=*/false);
    }

#pragma unroll
    for (int i = 0; i < 8; ++i)
        C[(size_t)(mBase + i + 8 * lg) * N + nBase + lm] = acc[i] * scale;
}

// ---------------------------------------------------------------------------
// Attention: one block per (b,h). Scores over L=2048 keys (stock / new / pad),
// mask, softmax (LDS reductions), attn written to d_out, out = attn @ V.
// Bandwidth-critical: K rows streamed with float4 (256B/row, fully consumed),
// V accumulation coalesced across d.
// ---------------------------------------------------------------------------
__global__ __launch_bounds__(256) void attn_kernel(const float* __restrict__ qh,   // (B, H*64), pre-scaled 1/8
                                                   const float* __restrict__ kn,   // (B, H*64)
                                                   const float* __restrict__ vn,   // (B, H*64)
                                                   const float* __restrict__ k_stock,
                                                   const float* __restrict__ v_stock,
                                                   const int*   __restrict__ mask, // (B, idx+1, L)
                                                   const int*   __restrict__ index_p,
                                                   float* __restrict__ attn_out,   // (B*H, L)
                                                   float* __restrict__ head_out)   // (B, H*64)
{
    constexpr int H = 8, DK = 64, L = 2048, DM = 512;
    const int bh  = blockIdx.x;          // 0..255
    const int b   = bh >> 3;
    const int h   = bh & 7;
    const int tid = threadIdx.x;
    const int idx = *index_p;            // 1024

    __shared__ float qv[DK];
    __shared__ float sc[L];
    __shared__ float red[256];
    __shared__ float part[256];

    if (tid < DK) qv[tid] = qh[(size_t)b * DM + h * DK + tid];
    __syncthreads();

    const float* krows = k_stock + (size_t)bh * idx * DK;
    const int*   mrow  = mask + ((size_t)b * (idx + 1) + idx) * L;

    float tmax = -INFINITY;
    for (int j = tid; j < L; j += 256) {
        float s;
        if (j < idx) {
            const float4* kr = (const float4*)(krows + (size_t)j * DK);
            float acc = 0.f;
#pragma unroll
            for (int d = 0; d < DK / 4; ++d) {
                float4 kk = kr[d];
                acc += kk.x * qv[4 * d + 0] + kk.y * qv[4 * d + 1]
                     + kk.z * qv[4 * d + 2] + kk.w * qv[4 * d + 3];
            }
            s = acc;
        } else if (j == idx) {
            const float* kr = kn + (size_t)b * DM + h * DK;
            float acc = 0.f;
#pragma unroll
            for (int d = 0; d < DK; ++d) acc += kr[d] * qv[d];
            s = acc;
        } else {
            s = 0.f;                      // zero-padded keys: score 0 pre-mask
        }
        if (mrow[j] == 0) s = NEG_INF_F;
        sc[j] = s;
        tmax  = fmaxf(tmax, s);
    }

    // block max
    red[tid] = tmax;
    __syncthreads();
    for (int st = 128; st > 0; st >>= 1) {
        if (tid < st) red[tid] = fmaxf(red[tid], red[tid + st]);
        __syncthreads();
    }
    const float mx = red[0];
    __syncthreads();

    float tsum = 0.f;
    for (int j = tid; j < L; j += 256) {
        float e = __expf(sc[j] - mx);
        sc[j] = e;
        tsum += e;
    }
    red[tid] = tsum;
    __syncthreads();
    for (int st = 128; st > 0; st >>= 1) {
        if (tid < st) red[tid] += red[tid + st];
        __syncthreads();
    }
    const float inv = 1.0f / red[0];
    __syncthreads();

    float* arow = attn_out + (size_t)bh * L;
    for (int j = tid; j < L; j += 256) {
        float a = sc[j] * inv;
        sc[j]   = a;
        arow[j] = a;
    }
    __syncthreads();

    // out[d] = sum_{j<=idx} attn[j] * V[j][d]; padded V rows are zero -> skipped
    const int d     = tid & 63;
    const int chunk = tid >> 6;          // 4 key-interleaved chunks
    const float* vrows = v_stock + (size_t)bh * idx * DK;
    float acc = 0.f;
    for (int j = chunk; j < idx; j += 4)
        acc += sc[j] * vrows[(size_t)j * DK + d];
    if (chunk == 0)
        acc += sc[idx] * vn[(size_t)b * DM + h * DK + d];
    part[tid] = acc;
    __syncthreads();
    if (tid < 64)
        head_out[(size_t)bh * DK + tid] =
            part[tid] + part[64 + tid] + part[128 + tid] + part[192 + tid];
}

// ---------------------------------------------------------------------------
// LayerNorm(fcout) + residual(q) -> out. One block per batch row of 512.
// ---------------------------------------------------------------------------
__global__ __launch_bounds__(256) void ln_kernel(const float* __restrict__ x,
                                                 const float* __restrict__ qres,
                                                 const float* __restrict__ g,
                                                 const float* __restrict__ bb,
                                                 float* __restrict__ out)
{
    constexpr int DM = 512;
    const int b   = blockIdx.x;
    const int tid = threadIdx.x;
    __shared__ float red[256];
    __shared__ float red2[256];

    const float* xr = x + (size_t)b * DM;
    float s = 0.f, s2 = 0.f;
    for (int i = tid; i < DM; i += 256) {
        float v = xr[i];
        s  += v;
        s2 += v * v;
    }
    red[tid]  = s;
    red2[tid] = s2;
    __syncthreads();
    for (int st = 128; st > 0; st >>= 1) {
        if (tid < st) {
            red[tid]  += red[tid + st];
            red2[tid] += red2[tid + st];
        }
        __syncthreads();
    }
    const float mu   = red[0]  * (1.0f / DM);
    const float var  = red2[0] * (1.0f / DM) - mu * mu;
    const float rinv = rsqrtf(var + 1e-6f);

    for (int i = tid; i < DM; i += 256)
        out[(size_t)b * DM + i] =
            (xr[i] - mu) * rinv * g[i] + bb[i] + qres[(size_t)b * DM + i];
}

// ---------------------------------------------------------------------------
extern "C" void kernel_launch(void* const* d_in, const int* in_sizes, int n_in,
                              void* d_out, int out_size, void* d_ws, size_t ws_size,
                              hipStream_t stream)
{
    const float* q       = (const float*)d_in[0];
    const float* k       = (const float*)d_in[1];
    const float* v       = (const float*)d_in[2];
    const float* k_stock = (const float*)d_in[3];
    const float* v_stock = (const float*)d_in[4];
    const int*   mask    = (const int*)  d_in[5];
    const float* w_qs    = (const float*)d_in[6];
    const float* w_ks    = (const float*)d_in[7];
    const float* w_vs    = (const float*)d_in[8];
    const float* w_fc    = (const float*)d_in[9];
    const float* ln_g    = (const float*)d_in[10];
    const float* ln_b    = (const float*)d_in[11];
    const int*   index_p = (const int*)  d_in[12];

    constexpr int B = 32, DM = 512, H = 8, L = 2048;
    float* ws      = (float*)d_ws;
    float* qh      = ws;                  // B*DM
    float* kn      = ws + 1 * B * DM;
    float* vn      = ws + 2 * B * DM;
    float* headout = ws + 3 * B * DM;
    float* fcout   = ws + 4 * B * DM;

    float* out  = (float*)d_out;          // (B, 1, DM)
    float* attn = out + B * DM;           // (B, H, 1, L)

    dim3 gblk(256);
    dim3 ggrid(4, 2);                     // 512/(8*16) N-tiles x 32/16 M-tiles

    // projections (q pre-scaled by 1/sqrt(dk) = 0.125)
    gemm32_wmma<<<ggrid, gblk, 0, stream>>>(q, w_qs, qh, 0.125f);
    gemm32_wmma<<<ggrid, gblk, 0, stream>>>(k, w_ks, kn, 1.0f);
    gemm32_wmma<<<ggrid, gblk, 0, stream>>>(v, w_vs, vn, 1.0f);

    // attention (bandwidth-bound: streams 134MB of K/V once)
    attn_kernel<<<B * H, 256, 0, stream>>>(qh, kn, vn, k_stock, v_stock,
                                           mask, index_p, attn, headout);

    // output projection + layernorm + residual
    gemm32_wmma<<<ggrid, gblk, 0, stream>>>(headout, w_fc, fcout, 1.0f);
    ln_kernel<<<B, 256, 0, stream>>>(fcout, q, ln_g, ln_b, out);
}